// GridSamplePScan_91001767067909
// MI455X (gfx1250) — compile-verified
//
#include <hip/hip_runtime.h>

typedef __attribute__((ext_vector_type(2))) float v2f;
typedef __attribute__((ext_vector_type(8))) float v8f;

#define L_T 16
#define C_N 8
#define H_N 128
#define W_N 128
#define HW   (H_N * W_N)   /* 16384 */
#define PLANE (2 * HW)     /* 32768 floats per time step of flows/cum */

// ---------------------------------------------------------------------------
// Kernel 1: cum = TRIL(16x16) @ F,  F = flows viewed as [16][32768] f32.
// One wave produces one 16x16 tile of cum via 4 chained V_WMMA_F32_16X16X4_F32.
// Layouts per CDNA5 ISA §7.12.2:
//   A 16x4 f32 : lane m(0-15) -> A[m][0],A[m][1]; lane m+16 -> A[m][2],A[m][3]
//   B 4x16 f32 : v0 = rows {K, K+2} across lane halves? -> per C/D convention:
//                v0 holds K=kb*4+2*half, v1 holds K=kb*4+2*half+1, N = lane&15
//   C/D 16x16  : VGPR i -> M = i + 8*(lane>=16), N = lane&15
// ---------------------------------------------------------------------------
__global__ __launch_bounds__(256)
void cumsum_wmma(const float* __restrict__ F, float* __restrict__ cum)
{
    int gtid = blockIdx.x * blockDim.x + threadIdx.x;
    int wave = gtid >> 5;          // 2048 waves total
    int lane = gtid & 31;
    int col0 = wave << 4;          // 16 columns per wave, covers 32768 cols
    int n    = lane & 15;          // A row (M) and B/D column (N)
    int half = lane >> 4;          // which K half this lane holds

    v8f acc = {};
#pragma unroll
    for (int kb = 0; kb < 4; ++kb) {
        int kbase = kb * 4 + half * 2;            // first K index in this lane
        v2f a, b;
        a.x = (kbase     <= n) ? 1.0f : 0.0f;     // TRIL: A[m=n][kbase]
        a.y = (kbase + 1 <= n) ? 1.0f : 0.0f;     // TRIL: A[m=n][kbase+1]
        b.x = F[(size_t)kbase       * PLANE + col0 + n];
        b.y = F[(size_t)(kbase + 1) * PLANE + col0 + n];
        acc = __builtin_amdgcn_wmma_f32_16x16x4_f32(
                  false, a, false, b, (short)0, acc, false, false);
    }
#pragma unroll
    for (int i = 0; i < 8; ++i) {
        int row = i + half * 8;                   // D layout: M = i + 8*half
        cum[(size_t)row * PLANE + col0 + n] = acc[i];
    }
}

// ---------------------------------------------------------------------------
// Kernel 2: bilinear sampling + triangular accumulation.
// One thread owns out[t, 0..7, y, x].
// ---------------------------------------------------------------------------
__global__ __launch_bounds__(256)
void sample_accum(const float* __restrict__ flows,
                  const float* __restrict__ images,
                  const float* __restrict__ cum,
                  int use_cum,
                  float* __restrict__ out)
{
    int tid = blockIdx.x * blockDim.x + threadIdx.x;
    int x = tid & (W_N - 1);
    int y = (tid >> 7) & (H_N - 1);
    int t = tid >> 14;                 // 0..15
    int p = y * W_N + x;

    float cumx[L_T], cumy[L_T];
    if (use_cum) {
#pragma unroll
        for (int j = 0; j < L_T; ++j) {
            cumx[j] = cum[(size_t)j * PLANE + p];
            cumy[j] = cum[(size_t)j * PLANE + HW + p];
        }
    } else {
        float sx = 0.0f, sy = 0.0f;
#pragma unroll
        for (int j = 0; j < L_T; ++j) {
            sx += flows[(size_t)j * PLANE + p];
            sy += flows[(size_t)j * PLANE + HW + p];
            cumx[j] = sx;
            cumy[j] = sy;
        }
    }

    const float basex = (x + 0.5f) * (2.0f / W_N) - 1.0f;
    const float basey = (y + 0.5f) * (2.0f / H_N) - 1.0f;

    float acc[C_N];
#pragma unroll
    for (int c = 0; c < C_N; ++c) acc[c] = 0.0f;

    const float ctx = cumx[t], cty = cumy[t];

    for (int k = 0; k <= t; ++k) {
        // grid = base + (cum[t] - cum[k]); wrap x: remainder(gx+1, 2) - 1
        float gx = basex + (ctx - cumx[k]);
        float u  = gx + 1.0f;
        u  = u - 2.0f * floorf(u * 0.5f);
        gx = u - 1.0f;
        float gy = basey + (cty - cumy[k]);

        float ixf = (gx + 1.0f) * (W_N * 0.5f) - 0.5f;
        float iyf = (gy + 1.0f) * (H_N * 0.5f) - 0.5f;
        float x0f = floorf(ixf), y0f = floorf(iyf);
        float wx  = ixf - x0f,   wy  = iyf - y0f;
        float x1f = x0f + 1.0f,  y1f = y0f + 1.0f;

        // zeros padding (validity computed on float coords, like the reference)
        float vx0 = (x0f >= 0.0f && x0f < (float)W_N) ? 1.0f : 0.0f;
        float vx1 = (x1f >= 0.0f && x1f < (float)W_N) ? 1.0f : 0.0f;
        float vy0 = (y0f >= 0.0f && y0f < (float)H_N) ? 1.0f : 0.0f;
        float vy1 = (y1f >= 0.0f && y1f < (float)H_N) ? 1.0f : 0.0f;

        int x0 = min(max((int)x0f, 0), W_N - 1);
        int x1 = min(max((int)x1f, 0), W_N - 1);
        int y0 = min(max((int)y0f, 0), H_N - 1);
        int y1 = min(max((int)y1f, 0), H_N - 1);

        float w00 = (1.0f - wx) * (1.0f - wy) * vx0 * vy0;
        float w01 = wx * (1.0f - wy) * vx1 * vy0;
        float w10 = (1.0f - wx) * wy * vx0 * vy1;
        float w11 = wx * wy * vx1 * vy1;

        const float* img = images + (size_t)k * C_N * HW;
        int i00 = y0 * W_N + x0, i01 = y0 * W_N + x1;
        int i10 = y1 * W_N + x0, i11 = y1 * W_N + x1;

#pragma unroll
        for (int c = 0; c < C_N; ++c) {
            const float* ic = img + (size_t)c * HW;
            acc[c] = fmaf(w00, ic[i00],
                     fmaf(w01, ic[i01],
                     fmaf(w10, ic[i10],
                     fmaf(w11, ic[i11], acc[c]))));
        }
    }

#pragma unroll
    for (int c = 0; c < C_N; ++c)
        out[(size_t)(t * C_N + c) * HW + p] = acc[c];
}

extern "C" void kernel_launch(void* const* d_in, const int* in_sizes, int n_in,
                              void* d_out, int out_size, void* d_ws, size_t ws_size,
                              hipStream_t stream)
{
    const float* flows  = (const float*)d_in[0];  // [1,16,2,128,128] f32
    const float* images = (const float*)d_in[1];  // [1,16,8,128,128] f32
    float* out = (float*)d_out;                   // [1,16,8,128,128] f32
    float* cum = (float*)d_ws;

    const size_t need = (size_t)L_T * PLANE * sizeof(float);  // 2 MB for cum
    const int use_cum = (d_ws != nullptr && ws_size >= need) ? 1 : 0;

    if (use_cum) {
        // 2048 waves * 32 lanes = 65536 threads, 256/block -> 256 blocks
        cumsum_wmma<<<256, 256, 0, stream>>>(flows, cum);
    }

    // 16 * 128 * 128 = 262144 threads -> 1024 blocks of 256
    sample_accum<<<1024, 256, 0, stream>>>(flows, images,
                                           use_cum ? cum : flows,
                                           use_cum, out);
}